// Quantizer_46634754900683
// MI455X (gfx1250) — compile-verified
//
#include <hip/hip_runtime.h>

typedef __attribute__((ext_vector_type(2))) float v2f;
typedef __attribute__((ext_vector_type(4))) float v4f;
typedef __attribute__((ext_vector_type(8))) float v8f;

#define DIMS 64
#define KCODES 512
#define CB_STRIDE 68              // pad 64->68 floats: stride 4 banks -> conflict-free ds_load_b64
#define WAVES_PER_BLOCK 4
#define ROWS_PER_WAVE 16
#define ROWS_PER_BLOCK (WAVES_PER_BLOCK * ROWS_PER_WAVE)
#define THREADS (WAVES_PER_BLOCK * 32)

__global__ void vq_zero_kernel(float* acc) { acc[0] = 0.0f; }

__global__ __launch_bounds__(THREADS, 1)
void vq_main_kernel(const float* __restrict__ z,
                    const float* __restrict__ cbg,
                    float* __restrict__ qout,
                    float* __restrict__ idsout,
                    float* __restrict__ lossacc) {
    extern __shared__ char smem[];
    float* cb    = (float*)smem;                                   // 512*68 f32
    float* cnorm = cb + KCODES * CB_STRIDE;                        // 512 f32
    float* rval  = cnorm + KCODES;                                 // 4*256 f32
    int*   ridx  = (int*)(rval + WAVES_PER_BLOCK * 256);           // 4*256 i32
    int*   rowid = ridx + WAVES_PER_BLOCK * 256;                   // 64 i32
    float* lred  = (float*)(rowid + ROWS_PER_BLOCK);               // 1 f32

    const int tid = threadIdx.x;

    // ---- preload codebook into LDS (padded rows) ----
    {
        const v4f* src = (const v4f*)cbg;                          // 512*16 float4
        for (int i = tid; i < KCODES * (DIMS / 4); i += THREADS) {
            int code = i >> 4, chunk = i & 15;
            *(v4f*)(cb + code * CB_STRIDE + chunk * 4) = src[i];
        }
    }
    if (tid == 0) lred[0] = 0.0f;
    __syncthreads();

    for (int c = tid; c < KCODES; c += THREADS) {
        float s = 0.0f;
        const float* p = cb + c * CB_STRIDE;
        #pragma unroll 8
        for (int j = 0; j < DIMS; ++j) { float v = p[j]; s += v * v; }
        cnorm[c] = s;
    }
    __syncthreads();

    const int wave    = tid >> 5;
    const int lane    = tid & 31;
    const int l16     = lane & 15;
    const int khalf   = lane >> 4;           // 0: K={0,1}, 1: K={2,3} within each 4-step
    const int rowbase = blockIdx.x * ROWS_PER_BLOCK + wave * ROWS_PER_WAVE;

    // ---- A fragments: whole 16x64 z tile in registers (16 frags x v2f) ----
    v2f a[16];
    {
        const float* zr = z + (size_t)(rowbase + l16) * DIMS + khalf * 2;
        #pragma unroll
        for (int j = 0; j < 16; ++j)
            a[j] = *(const v2f*)(zr + j * 4);
    }

    float minv[8];
    int   mini[8];
    #pragma unroll
    for (int r = 0; r < 8; ++r) { minv[r] = 3.4e38f; mini[r] = 0; }

    // ---- 32 code tiles x 16 K-steps of fp32 WMMA ----
    for (int t = 0; t < 32; ++t) {
        const int code = t * 16 + l16;       // this lane's column (N) index
        const float* bp = cb + code * CB_STRIDE + khalf * 2;
        v8f acc = {};
        #pragma unroll
        for (int j = 0; j < 16; ++j) {
            v2f b = *(const v2f*)(bp + j * 4);
            acc = __builtin_amdgcn_wmma_f32_16x16x4_f32(
                false, a[j], false, b, (short)0, acc, false, false);
        }
        const float cn = cnorm[code];
        #pragma unroll
        for (int r = 0; r < 8; ++r) {
            float cand = cn - 2.0f * acc[r]; // ||c||^2 - 2 z.c  (||z||^2 common per row)
            if (cand < minv[r]) { minv[r] = cand; mini[r] = code; }
        }
    }

    // ---- per-row argmin across the 16 lanes holding that row ----
    #pragma unroll
    for (int r = 0; r < 8; ++r) {
        int rr = r + 8 * khalf;              // C-layout: VGPR r holds rows r (lanes 0-15) / r+8 (16-31)
        rval[wave * 256 + rr * 16 + l16] = minv[r];
        ridx[wave * 256 + rr * 16 + l16] = mini[r];
    }
    __syncthreads();

    if (lane < 16) {
        float bv = 3.4e38f; int bi = 0x7fffffff;
        for (int c = 0; c < 16; ++c) {
            float v = rval[wave * 256 + lane * 16 + c];
            int   i = ridx[wave * 256 + lane * 16 + c];
            if (v < bv || (v == bv && i < bi)) { bv = v; bi = i; }
        }
        rowid[wave * 16 + lane] = bi;
        idsout[rowbase + lane] = (float)bi;
    }
    __syncthreads();

    // ---- gather q, write out, accumulate loss (coalesced over dims) ----
    float lsum = 0.0f;
    for (int r = 0; r < 16; ++r) {
        int id = rowid[wave * 16 + r];
        size_t grow = (size_t)(rowbase + r) * DIMS;
        float q0 = cb[id * CB_STRIDE + lane];
        float q1 = cb[id * CB_STRIDE + 32 + lane];
        float z0 = z[grow + lane];
        float z1 = z[grow + 32 + lane];
        qout[grow + lane]      = q0;
        qout[grow + 32 + lane] = q1;
        float d0 = z0 - q0, d1 = z1 - q1;
        lsum += d0 * d0 + d1 * d1;
    }
    atomicAdd(lred, lsum);                   // ds_add_f32
    __syncthreads();
    if (tid == 0) atomicAdd(lossacc, lred[0]);
}

__global__ void vq_final_kernel(const float* acc, float* out, float invND) {
    out[0] = 1.25f * acc[0] * invND;         // both loss terms value-equal
}

extern "C" void kernel_launch(void* const* d_in, const int* in_sizes, int n_in,
                              void* d_out, int out_size, void* d_ws, size_t ws_size,
                              hipStream_t stream) {
    const float* z   = (const float*)d_in[0];
    const float* cbg = (const float*)d_in[1];
    const int N = in_sizes[0] / DIMS;        // 262144

    float* qout    = (float*)d_out;
    float* lossout = qout + (size_t)N * DIMS;
    float* idsout  = lossout + 1;
    float* acc     = (float*)d_ws;

    size_t smem = (size_t)(KCODES * CB_STRIDE + KCODES + WAVES_PER_BLOCK * 256) * sizeof(float)
                + (size_t)(WAVES_PER_BLOCK * 256 + ROWS_PER_BLOCK) * sizeof(int)
                + sizeof(float);
    (void)hipFuncSetAttribute((const void*)vq_main_kernel,
                              hipFuncAttributeMaxDynamicSharedMemorySize, (int)smem);

    vq_zero_kernel<<<1, 1, 0, stream>>>(acc);
    vq_main_kernel<<<N / ROWS_PER_BLOCK, THREADS, smem, stream>>>(z, cbg, qout, idsout, acc);
    vq_final_kernel<<<1, 1, 0, stream>>>(acc, lossout, 1.0f / (float)((size_t)N * DIMS));
}